// GraphConvolution_19662360281445
// MI455X (gfx1250) — compile-verified
//
#include <hip/hip_runtime.h>
#include <hip/hip_bf16.h>

// GraphConvolution forward for MI455X (gfx1250, wave32):
//   support = x @ W          -> fp32 WMMA (V_WMMA_F32_16X16X4_F32), 1 wave per 16x64 tile
//   out     = scatter-add of edge_val * support[col] into rows     -> global_atomic_add_f32
//   out     = relu(out)      -> vectorized epilogue pass
//
// All phases are memory-bound: x is 25.6MB, edge lists 16MB, support/out are
// 12.8MB each and fully L2-resident (192MB L2), so the gather/atomic phase runs
// at L2 bandwidth. fp32 WMMA keeps reference precision at zero cost since the
// GEMM is bandwidth-limited anyway.

#define N_NODES 50000
#define N_EDGES 800000
#define IN_DIM  128
#define OUT_DIM 64

typedef float v2f __attribute__((ext_vector_type(2)));
typedef float v8f __attribute__((ext_vector_type(8)));

// ---------------------------------------------------------------------------
// Kernel 1: support[N_NODES, 64] = x[N_NODES, 128] @ W[128, 64]
// One wave -> 16 rows x 64 cols. 4 f32 accumulators (16x16 each), K in steps
// of 4 via v_wmma_f32_16x16x4_f32 (A reused across the 4 N-tiles).
// ---------------------------------------------------------------------------
__global__ __launch_bounds__(128)
void gcn_gemm_wmma(const float* __restrict__ x,
                   const float* __restrict__ w,
                   float* __restrict__ support) {
    const int lane = threadIdx.x & 31;
    const int wave = (blockIdx.x * (blockDim.x >> 5)) + (threadIdx.x >> 5);
    const int row0 = wave * 16;
    if (row0 >= N_NODES) return;

    const int m  = lane & 15;          // A: M row / B: N col / C: N col (per half-wave)
    const int kh = (lane >> 4) << 1;   // K sub-pair: 0 for lanes 0-15, 2 for lanes 16-31

    // A layout (ISA 7.12.2, 32-bit A 16x4): lane<16 -> {K=0,K=1}, lane>=16 -> {K=2,K=3}
    const float* arow = x + (size_t)(row0 + m) * IN_DIM + kh;

    v8f acc0 = {}, acc1 = {}, acc2 = {}, acc3 = {};

    #pragma unroll 4
    for (int k = 0; k < IN_DIM; k += 4) {
        v2f a = *(const v2f*)(arow + k);              // global_load_b64, 8B aligned

        // B layout mirrors A: lane holds column (N = j*16 + m), K-pair = kh
        const float* wk = w + (size_t)(k + kh) * OUT_DIM + m;
        v2f b0, b1, b2, b3;
        b0.x = wk[0];   b0.y = wk[OUT_DIM + 0];
        b1.x = wk[16];  b1.y = wk[OUT_DIM + 16];
        b2.x = wk[32];  b2.y = wk[OUT_DIM + 32];
        b3.x = wk[48];  b3.y = wk[OUT_DIM + 48];

        acc0 = __builtin_amdgcn_wmma_f32_16x16x4_f32(false, a, false, b0,
                                                     (short)0, acc0, false, false);
        acc1 = __builtin_amdgcn_wmma_f32_16x16x4_f32(false, a, false, b1,
                                                     (short)0, acc1, false, false);
        acc2 = __builtin_amdgcn_wmma_f32_16x16x4_f32(false, a, false, b2,
                                                     (short)0, acc2, false, false);
        acc3 = __builtin_amdgcn_wmma_f32_16x16x4_f32(false, a, false, b3,
                                                     (short)0, acc3, false, false);
    }

    // C/D layout: VGPR r holds M = r + 8*(lane>>4), N = lane&15
    const int mh = (lane >> 4) * 8;
    #pragma unroll
    for (int r = 0; r < 8; ++r) {
        const size_t orow = (size_t)(row0 + mh + r) * OUT_DIM;
        support[orow +  0 + m] = acc0[r];
        support[orow + 16 + m] = acc1[r];
        support[orow + 32 + m] = acc2[r];
        support[orow + 48 + m] = acc3[r];
    }
}

// ---------------------------------------------------------------------------
// Kernel 2: out[row[e], :] += val[e] * support[col[e], :]
// Thread t -> (edge = t/64, feat = t%64). Each wave covers 32 contiguous
// features of one edge: coalesced 128B gather from L2-resident support and
// coalesced fp32 atomics into the L2-resident output.
// ---------------------------------------------------------------------------
__global__ __launch_bounds__(256)
void gcn_scatter(const long long* __restrict__ erow,
                 const long long* __restrict__ ecol,
                 const float* __restrict__ eval,
                 const float* __restrict__ support,
                 float* __restrict__ out) {
    const long long t = (long long)blockIdx.x * 256 + threadIdx.x;
    const int e = (int)(t >> 6);
    const int f = (int)(t & 63);
    if (e >= N_EDGES) return;

    const int r = (int)erow[e];      // uniform across the wave -> scalarized
    const int c = (int)ecol[e];
    const float v = eval[e];

    const float msg = v * support[(size_t)c * OUT_DIM + f];
    atomicAdd(out + (size_t)r * OUT_DIM + f, msg);
}

// ---------------------------------------------------------------------------
// Kernel 3: in-place ReLU, float4 vectorized (must run after all atomics).
// ---------------------------------------------------------------------------
__global__ __launch_bounds__(256)
void gcn_relu(float* __restrict__ out, int n4) {
    const int i = blockIdx.x * 256 + threadIdx.x;
    if (i < n4) {
        float4 vv = ((const float4*)out)[i];
        vv.x = fmaxf(vv.x, 0.0f);
        vv.y = fmaxf(vv.y, 0.0f);
        vv.z = fmaxf(vv.z, 0.0f);
        vv.w = fmaxf(vv.w, 0.0f);
        ((float4*)out)[i] = vv;
    }
}

extern "C" void kernel_launch(void* const* d_in, const int* in_sizes, int n_in,
                              void* d_out, int out_size, void* d_ws, size_t ws_size,
                              hipStream_t stream) {
    (void)in_sizes; (void)n_in; (void)ws_size;

    const float*     x    = (const float*)d_in[0];
    const long long* erow = (const long long*)d_in[1];   // int64 per reference
    const long long* ecol = (const long long*)d_in[2];   // int64 per reference
    const float*     ev   = (const float*)d_in[3];
    const float*     w    = (const float*)d_in[4];
    float*           out  = (float*)d_out;
    float*           supp = (float*)d_ws;                // 50000*64*4 = 12.8 MB scratch

    // Zero the atomic accumulation target (graph-capture-safe stream op).
    hipMemsetAsync(d_out, 0, (size_t)out_size * sizeof(float), stream);

    // GEMM: 3125 waves of 32, 4 waves per block.
    const int n_waves  = N_NODES / 16;                   // 3125
    const int g_blocks = (n_waves + 3) / 4;              // 782
    gcn_gemm_wmma<<<g_blocks, 128, 0, stream>>>(x, w, supp);

    // Scatter-add: 800000 edges * 64 feats / 256 threads.
    const long long s_threads = (long long)N_EDGES * OUT_DIM;
    const int s_blocks = (int)((s_threads + 255) / 256); // 200000
    gcn_scatter<<<s_blocks, 256, 0, stream>>>(erow, ecol, ev, supp, out);

    // ReLU epilogue.
    const int n4 = (N_NODES * OUT_DIM) / 4;              // 800000
    gcn_relu<<<(n4 + 255) / 256, 256, 0, stream>>>(out, n4);
}